// Graph_cell_25125558682340
// MI455X (gfx1250) — compile-verified
//
#include <hip/hip_runtime.h>
#include <hip/hip_bf16.h>
#include <stdint.h>

// GCN layer: out = ((A@X) * inv_colsum(A)) @ Wn + X @ Ws + bias
// B=4, N=4096, C=64, f32.  HBM floor = one pass over A (256MB) ~ 11us.
//
// k1 (main): per (b, 16-row block), double-buffered async-to-LDS staging of
//   A (16 x 64 k-chunk) and X (64 x 64 k-chunk) via global_load_async_to_lds_b128
//   (ASYNCcnt), WMMA f32 16x16x4 from LDS, column sums computed from the
//   LDS-resident A tile (atomicAdd into deg). Raw message parked in d_out.
// k2: scale by 1/deg, msg@Wn + X@Ws + bias via 32 more WMMAs per wave.

#define NN 4096
#define CC 64

typedef __attribute__((ext_vector_type(2))) float v2f;
typedef __attribute__((ext_vector_type(8))) float v8f;

__device__ __forceinline__ void async_b128(uint32_t lds_byte, uint32_t voff_byte,
                                           uint64_t sbase) {
    // GVS form: mem = SADDR + VADDR(voff) ; dest = LDS byte address in VGPR
    asm volatile("global_load_async_to_lds_b128 %0, %1, %2"
                 :: "v"(lds_byte), "v"(voff_byte), "s"(sbase)
                 : "memory");
}

__global__ void gcn_zero_deg(float* __restrict__ deg) {
    deg[blockIdx.x * 256 + threadIdx.x] = 0.0f;
}

__global__ __launch_bounds__(128) void gcn_msg_kernel(
    const float* __restrict__ X,   // [4][4096][64]
    const float* __restrict__ A,   // [4][4096][4096]
    float* __restrict__ msg,       // [4][4096][64]  (= d_out, raw message)
    float* __restrict__ deg)       // [4][4096]
{
    constexpr int KC   = 64;        // k-chunk
    constexpr int NCH  = NN / KC;   // 64 chunks
    constexpr int AST  = 68;        // A LDS row stride (floats): banks (4m+k)%64
    constexpr int XST  = 80;        // X LDS row stride (floats): banks (16k+c)%64
    constexpr int ABUF = 16 * AST;  // floats per A buffer
    constexpr int XBUF = KC * XST;  // floats per X buffer

    __shared__ float sA[2 * ABUF];  // ~8.5 KB
    __shared__ float sX[2 * XBUF];  // ~40 KB

    const int blk      = blockIdx.x;
    const int b        = blk >> 8;
    const int row_base = (blk & 255) * 16;
    const int tid      = threadIdx.x;
    const int wave     = tid >> 5;
    const int lane     = tid & 31;
    const int m        = lane & 15;
    const int half     = lane >> 4;
    const int koff     = half * 2;
    const int col      = wave * 16 + m;

    const uint64_t Ag = (uint64_t)(uintptr_t)(A + (size_t)b * NN * NN
                                                + (size_t)row_base * NN);
    const uint64_t Xg = (uint64_t)(uintptr_t)(X + (size_t)b * NN * CC);
    const uint32_t sAb = (uint32_t)(uintptr_t)&sA[0];
    const uint32_t sXb = (uint32_t)(uintptr_t)&sX[0];

    // Per-thread async transfer descriptors (16B granules).
    // A tile: 16 rows x 16 granules = 256 -> 2 per thread.
    // X tile: 64 rows x 16 granules = 1024 -> 8 per thread.
    uint32_t aV[2], aL[2], xV[8], xL[8];
#pragma unroll
    for (int i = 0; i < 2; ++i) {
        int cid = tid + 128 * i, r = cid >> 4, s = cid & 15;
        aV[i] = ((uint32_t)r * NN + s * 4) * 4;   // global byte offset (k=0)
        aL[i] = (uint32_t)(r * AST + s * 4) * 4;  // LDS byte offset in buffer
    }
#pragma unroll
    for (int j = 0; j < 8; ++j) {
        int cid = tid + 128 * j, kr = cid >> 4, s = cid & 15;
        xV[j] = ((uint32_t)kr * CC + s * 4) * 4;
        xL[j] = (uint32_t)(kr * XST + s * 4) * 4;
    }

    auto issue = [&](int chunk, int buf) {
        uint32_t av = (uint32_t)chunk * KC * 4;        // k advances along A rows
        uint32_t xv = (uint32_t)chunk * KC * CC * 4;   // k advances down X rows
        uint32_t ab = sAb + (uint32_t)buf * ABUF * 4;
        uint32_t xb = sXb + (uint32_t)buf * XBUF * 4;
#pragma unroll
        for (int i = 0; i < 2; ++i) async_b128(ab + aL[i], aV[i] + av, Ag);
#pragma unroll
        for (int j = 0; j < 8; ++j) async_b128(xb + xL[j], xV[j] + xv, Xg);
    };

    v8f acc = {};
    issue(0, 0);
    for (int c = 0; c < NCH; ++c) {
        const int buf = c & 1;
        if (c + 1 < NCH) {
            issue(c + 1, buf ^ 1);                       // prefetch next chunk
            asm volatile("s_wait_asynccnt 0xa" ::: "memory");  // this chunk done
        } else {
            asm volatile("s_wait_asynccnt 0x0" ::: "memory");
        }
        __syncthreads();

        const float* __restrict__ Ab = sA + buf * ABUF;
        const float* __restrict__ Xb = sX + buf * XBUF;
#pragma unroll
        for (int kk = 0; kk < KC; kk += 4) {
            v2f a = *(const v2f*)(Ab + m * AST + kk + koff);
            v2f bv;
            bv.x = Xb[(kk + koff) * XST + col];
            bv.y = Xb[(kk + koff + 1) * XST + col];
            acc = __builtin_amdgcn_wmma_f32_16x16x4_f32(
                false, a, false, bv, (short)0, acc, false, false);
        }

        // Column sums of this A tile straight from LDS (16 rows each).
        if (tid < KC) {
            float s = 0.0f;
#pragma unroll
            for (int r = 0; r < 16; ++r) s += Ab[r * AST + tid];
            atomicAdd(&deg[b * NN + c * KC + tid], s);
        }
        __syncthreads();   // reads done before this buffer is re-filled
    }

    // Park raw message tile in d_out: D-VGPR r -> row (r + 8*half)
    float* __restrict__ ob = msg + (size_t)b * NN * CC;
#pragma unroll
    for (int r = 0; r < 8; ++r) {
        int row = row_base + r + 8 * half;
        ob[(size_t)row * CC + col] = acc[r];
    }
}

__global__ __launch_bounds__(128) void gcn_final_kernel(
    const float* __restrict__ X,    // [4][4096][64]
    const float* __restrict__ Wn,   // [64][64]
    const float* __restrict__ Ws,   // [64][64]
    const float* __restrict__ bias, // [64]
    float* __restrict__ out,        // d_out: raw msg in, final out
    const float* __restrict__ deg)  // [4][4096]
{
    __shared__ float msgS[16][CC + 1];  // +1 pad: conflict-free

    const int blk      = blockIdx.x;
    const int b        = blk >> 8;
    const int row_base = (blk & 255) * 16;
    const int tid      = threadIdx.x;
    const int wave     = tid >> 5;
    const int lane     = tid & 31;
    const int m        = lane & 15;
    const int half     = lane >> 4;
    const int koff     = half * 2;
    const int col      = wave * 16 + m;

    float* __restrict__ ob = out + (size_t)b * NN * CC;

    // 1) load raw message tile, scale by 1/deg, stage in LDS
    for (int idx = tid; idx < 16 * CC; idx += 128) {
        int r = idx >> 6;
        int c = idx & 63;
        float d   = deg[b * NN + row_base + r];
        float inv = (d != 0.0f) ? (1.0f / d) : 0.0f;
        msgS[r][c] = ob[(size_t)(row_base + r) * CC + c] * inv;
    }
    __syncthreads();

    // 2) acc = msgS @ Wn + Xtile @ Ws   (K = 64, two WMMAs per k-step)
    const float* __restrict__ Xrow =
        X + ((size_t)b * NN + (row_base + m)) * CC + koff;

    v8f acc = {};
#pragma unroll
    for (int k = 0; k < CC; k += 4) {
        v2f a1;
        a1.x = msgS[m][k + koff];
        a1.y = msgS[m][k + koff + 1];
        v2f b1;
        b1.x = Wn[(k + koff) * CC + col];
        b1.y = Wn[(k + koff + 1) * CC + col];
        acc = __builtin_amdgcn_wmma_f32_16x16x4_f32(
            false, a1, false, b1, (short)0, acc, false, false);

        v2f a2 = *(const v2f*)(Xrow + k);
        v2f b2;
        b2.x = Ws[(k + koff) * CC + col];
        b2.y = Ws[(k + koff + 1) * CC + col];
        acc = __builtin_amdgcn_wmma_f32_16x16x4_f32(
            false, a2, false, b2, (short)0, acc, false, false);
    }

    // 3) + bias, overwrite our rows of d_out
    float bv = bias[col];
#pragma unroll
    for (int r = 0; r < 8; ++r) {
        int row = row_base + r + 8 * half;
        ob[(size_t)row * CC + col] = acc[r] + bv;
    }
}

extern "C" void kernel_launch(void* const* d_in, const int* in_sizes, int n_in,
                              void* d_out, int out_size, void* d_ws, size_t ws_size,
                              hipStream_t stream) {
    const float* X    = (const float*)d_in[0];
    const float* A    = (const float*)d_in[1];
    const float* Wn   = (const float*)d_in[2];
    const float* Ws   = (const float*)d_in[3];
    const float* bias = (const float*)d_in[4];
    float* out = (float*)d_out;
    float* deg = (float*)d_ws;   // 4*4096 floats = 64 KB

    const int B = 4;
    gcn_zero_deg<<<B * NN / 256, 256, 0, stream>>>(deg);
    gcn_msg_kernel<<<B * (NN / 16), 128, 0, stream>>>(X, A, out, deg);
    gcn_final_kernel<<<B * (NN / 16), 128, 0, stream>>>(X, Wn, Ws, bias, out, deg);
}